// LSTM_16647293240166
// MI455X (gfx1250) — compile-verified
//
#include <hip/hip_runtime.h>
#include <cmath>
#include <cstdint>

// ---------------- problem constants ----------------
#define S_LEN   1024
#define BATCH   64
#define ISZ     512
#define HSZ     512

// ---------------- kernel geometry ------------------
#define NWG     64          // persistent workgroups, one per 8 hidden units
#define HPW     8           // hidden units per WG
#define NCOL    32          // gate columns per WG = 4 gates * HPW
#define NTHR    256         // 8 wave32 per WG: 4 batch-tiles x 2 col-tiles
#define KC      128         // A-chunk depth staged in LDS (double buffered)
#define WSTRIDE 1028        // 1024 + 4 pad  -> conflict-free b64 LDS reads
#define ASTRIDE 132         // 128 + 4 pad   -> conflict-free b64 LDS reads
#define GSTRIDE 33          // gate buffer row stride

typedef float v2f __attribute__((ext_vector_type(2)));
typedef float v8f __attribute__((ext_vector_type(8)));

__device__ __forceinline__ float sigmoidf(float v) { return 1.0f / (1.0f + expf(-v)); }

// ---- CDNA5 async global->LDS copy path (ASYNCcnt), with safe fallback ----
#if defined(__has_builtin)
#if __has_builtin(__builtin_amdgcn_global_load_async_to_lds_b128)
#define HAVE_ASYNC_LDS 1
#endif
#endif

#if defined(HAVE_ASYNC_LDS)
// builtin expects: (v4i32 addrspace(1)*, v4i32 addrspace(3)*, imm offset, imm cpol)
typedef int v4i __attribute__((vector_size(16)));
typedef __attribute__((address_space(1))) v4i* gv4i_ptr;
typedef __attribute__((address_space(3))) v4i* lv4i_ptr;

__device__ __forceinline__ void async_copy16(const float* g, float* l) {
  // generic LDS pointers keep the LDS byte offset in their low 32 bits
  __builtin_amdgcn_global_load_async_to_lds_b128(
      (gv4i_ptr)(uintptr_t)g, (lv4i_ptr)(uint32_t)(uintptr_t)l, 0, 0);
}
__device__ __forceinline__ void wait_async0() {
#if __has_builtin(__builtin_amdgcn_s_wait_asynccnt)
  __builtin_amdgcn_s_wait_asynccnt(0);
#else
  asm volatile("s_wait_asynccnt 0x0" ::: "memory");
#endif
}
#endif

// Persistent LSTM: weights LDS-resident for all 1024 steps; one grid barrier
// per step; h double-buffered in global scratch (L2-hot, 128 KB).
__global__ __launch_bounds__(NTHR, 1)
void lstm_persistent_kernel(const float* __restrict__ x,
                            const float* __restrict__ h0,
                            const float* __restrict__ c0,
                            const float* __restrict__ Wi,
                            const float* __restrict__ bi,
                            const float* __restrict__ Wh,
                            const float* __restrict__ bh,
                            float* __restrict__ out,
                            float* __restrict__ hbuf,    // 2 * BATCH*HSZ floats
                            unsigned* __restrict__ cnt)  // grid-barrier counter
{
  extern __shared__ float smem[];
  float* Wl = smem;                          // NCOL * WSTRIDE   (fused [Wi|Wh] slice)
  float* Al = Wl + NCOL * WSTRIDE;           // 2 * BATCH * ASTRIDE (A chunks, ping-pong)
  float* Gl = Al + 2 * BATCH * ASTRIDE;      // BATCH * GSTRIDE  (gate tiles)

  const int tid  = threadIdx.x;
  const int lane = tid & 31;
  const int wv   = tid >> 5;
  const int m    = wv & 3;                   // batch tile (rows m*16..m*16+15)
  const int n    = wv >> 2;                  // column tile (cols n*16..n*16+15)
  const int r    = lane & 15;
  const int koff = (lane >> 4) << 1;         // K sub-offset per WMMA A/B layout: 0 or 2
  const int hb   = blockIdx.x * HPW;         // first hidden unit owned by this WG

  // ---- stage fused [Wi | Wh] slice into LDS (once per launch) ----
  // column c (0..31): gate q = c>>3 (f,i,g,o), hidden j = c&7, global row q*512+hb+j
  for (int i = 0; i < 32; ++i) {
    int e4 = tid + i * NTHR;                 // 8192 float4 elements total
    int c  = e4 >> 8;
    int k4 = (e4 & 255) << 2;                // 0..1020
    int R  = ((c >> 3) * HSZ) + hb + (c & 7);
    const float* src = (k4 < ISZ) ? (Wi + (size_t)R * ISZ + k4)
                                  : (Wh + (size_t)R * HSZ + (k4 - ISZ));
    *(float4*)(Wl + c * WSTRIDE + k4) = *(const float4*)src;
  }

  // per-lane folded bias (bi + bh) for this wave's output column
  const int cc  = n * 16 + r;                // 0..31 within WG
  const int Rcc = ((cc >> 3) * HSZ) + hb + (cc & 7);
  const float bias = bi[Rcc] + bh[Rcc];

  // epilogue mapping: 512 (batch, hidden) pairs per WG -> 2 per thread
  const int j0 = tid & 7;
  const int b0 = tid >> 3;                   // 0..31
  const int b1 = b0 + 32;                    // 32..63
  float cr0 = c0[(size_t)b0 * HSZ + hb + j0];
  float cr1 = c0[(size_t)b1 * HSZ + hb + j0];
  hbuf[(size_t)b0 * HSZ + hb + j0] = h0[(size_t)b0 * HSZ + hb + j0];
  hbuf[(size_t)b1 * HSZ + hb + j0] = h0[(size_t)b1 * HSZ + hb + j0];

  float* outH = out + (size_t)S_LEN * BATCH * HSZ;
  float* outC = outH + (size_t)BATCH * HSZ;

  const float* Wlane    = Wl + cc * WSTRIDE + koff;
  const int    arow_off = (m * 16 + r) * ASTRIDE + koff;

  for (int t = 0; t < S_LEN; ++t) {
    const float* xt    = x    + (size_t)t * (BATCH * ISZ);
    const float* hcur  = hbuf + (size_t)(t & 1) * (BATCH * HSZ);
    float*       hnext = hbuf + (size_t)((t + 1) & 1) * (BATCH * HSZ);

#if defined(HAVE_ASYNC_LDS)
    // issue chunk 0 (pure x data) as async global->LDS before the barrier
#pragma unroll
    for (int i = 0; i < 8; ++i) {
      int q = tid + i * NTHR, b = q >> 5, k4 = (q & 31) << 2;
      async_copy16(xt + (size_t)b * ISZ + k4, Al + b * ASTRIDE + k4);
    }
#else
    float4 pf[8];
#pragma unroll
    for (int i = 0; i < 8; ++i) {
      int q = tid + i * NTHR, b = q >> 5, k4 = (q & 31) << 2;
      pf[i] = *(const float4*)(xt + (size_t)b * ISZ + k4);
    }
#endif

    // ---- device-wide barrier: makes h(t) visible to every WG ----
    __threadfence();
#if defined(HAVE_ASYNC_LDS)
    wait_async0();                 // own chunk-0 copies done before joining
#endif
    __syncthreads();
    if (tid == 0) {
      __hip_atomic_fetch_add(cnt, 1u, __ATOMIC_ACQ_REL, __HIP_MEMORY_SCOPE_AGENT);
      const unsigned tgt = (unsigned)(t + 1) * NWG;
      while (__hip_atomic_load(cnt, __ATOMIC_ACQUIRE, __HIP_MEMORY_SCOPE_AGENT) < tgt)
        __builtin_amdgcn_s_sleep(1);
    }
    __syncthreads();

#if !defined(HAVE_ASYNC_LDS)
    // commit chunk 0 to LDS (fallback path)
#pragma unroll
    for (int i = 0; i < 8; ++i) {
      int q = tid + i * NTHR, b = q >> 5, k4 = (q & 31) << 2;
      *(float4*)(Al + b * ASTRIDE + k4) = pf[i];
    }
    __syncthreads();
#endif

    v8f acc;
#pragma unroll
    for (int v = 0; v < 8; ++v) acc[v] = bias;

    // K = 1024 fused (x: chunks 0..3, h: chunks 4..7), double-buffered in LDS
    for (int kc = 0; kc < 8; ++kc) {
      const int cur = kc & 1;
      if (kc < 7) {                                   // stage next chunk
        const float* src = (kc < 3) ? (xt + (kc + 1) * KC)
                                    : (hcur + (kc - 3) * KC);
#if defined(HAVE_ASYNC_LDS)
        float* dst = Al + (cur ^ 1) * (BATCH * ASTRIDE);
#pragma unroll
        for (int i = 0; i < 8; ++i) {
          int q = tid + i * NTHR, b = q >> 5, k4 = (q & 31) << 2;
          async_copy16(src + (size_t)b * 512 + k4, dst + b * ASTRIDE + k4);
        }
#else
#pragma unroll
        for (int i = 0; i < 8; ++i) {
          int q = tid + i * NTHR, b = q >> 5, k4 = (q & 31) << 2;
          pf[i] = *(const float4*)(src + (size_t)b * 512 + k4);
        }
#endif
      }
      // WMMA over this chunk: A(16x4) from Al, B(4x16) = W^T slice from Wl
      const float* Ap = Al + cur * (BATCH * ASTRIDE) + arow_off;
      const float* Wp = Wlane + kc * KC;
#pragma unroll
      for (int k = 0; k < KC; k += 4) {
        v2f av = *(const v2f*)(Ap + k);               // ds_load_b64
        v2f bv = *(const v2f*)(Wp + k);               // ds_load_b64
        acc = __builtin_amdgcn_wmma_f32_16x16x4_f32(false, av, false, bv,
                                                    (short)0, acc, false, false);
      }
      if (kc < 7) {
#if defined(HAVE_ASYNC_LDS)
        wait_async0();                                // own async copies landed
#else
        float* dst = Al + (cur ^ 1) * (BATCH * ASTRIDE);
#pragma unroll
        for (int i = 0; i < 8; ++i) {
          int q = tid + i * NTHR, b = q >> 5, k4 = (q & 31) << 2;
          *(float4*)(dst + b * ASTRIDE + k4) = pf[i];
        }
#endif
      }
      __syncthreads();
    }

    // scatter C tiles to the gate buffer (D layout: vgpr v -> M = v + 8*(lane>=16))
    const int half8 = (lane >> 4) << 3;
#pragma unroll
    for (int v = 0; v < 8; ++v)
      Gl[(m * 16 + half8 + v) * GSTRIDE + cc] = acc[v];
    __syncthreads();

    // gate math + state update: cols j, 8+j, 16+j, 24+j = f,i,g,o for hidden j
    {
      float fg = sigmoidf(Gl[b0 * GSTRIDE +      j0]);
      float ig = sigmoidf(Gl[b0 * GSTRIDE +  8 + j0]);
      float gg = tanhf   (Gl[b0 * GSTRIDE + 16 + j0]);
      float og = sigmoidf(Gl[b0 * GSTRIDE + 24 + j0]);
      cr0 = fg * cr0 + ig * gg;
      float hn = og * tanhf(cr0);
      hnext[(size_t)b0 * HSZ + hb + j0] = hn;
      out[(size_t)t * (BATCH * HSZ) + (size_t)b0 * HSZ + hb + j0] = hn;
      if (t == S_LEN - 1) outH[(size_t)b0 * HSZ + hb + j0] = hn;
    }
    {
      float fg = sigmoidf(Gl[b1 * GSTRIDE +      j0]);
      float ig = sigmoidf(Gl[b1 * GSTRIDE +  8 + j0]);
      float gg = tanhf   (Gl[b1 * GSTRIDE + 16 + j0]);
      float og = sigmoidf(Gl[b1 * GSTRIDE + 24 + j0]);
      cr1 = fg * cr1 + ig * gg;
      float hn = og * tanhf(cr1);
      hnext[(size_t)b1 * HSZ + hb + j0] = hn;
      out[(size_t)t * (BATCH * HSZ) + (size_t)b1 * HSZ + hb + j0] = hn;
      if (t == S_LEN - 1) outH[(size_t)b1 * HSZ + hb + j0] = hn;
    }
  }

  // final cell state
  outC[(size_t)b0 * HSZ + hb + j0] = cr0;
  outC[(size_t)b1 * HSZ + hb + j0] = cr1;
}

__global__ void lstm_init_kernel(unsigned* cnt) {
  if (threadIdx.x == 0 && blockIdx.x == 0) *cnt = 0u;
}

extern "C" void kernel_launch(void* const* d_in, const int* in_sizes, int n_in,
                              void* d_out, int out_size, void* d_ws, size_t ws_size,
                              hipStream_t stream) {
  const float* x  = (const float*)d_in[0];
  const float* h0 = (const float*)d_in[1];
  const float* c0 = (const float*)d_in[2];
  const float* Wi = (const float*)d_in[3];
  const float* bi = (const float*)d_in[4];
  const float* Wh = (const float*)d_in[5];
  const float* bh = (const float*)d_in[6];
  float* out = (float*)d_out;

  unsigned* cnt = (unsigned*)d_ws;                        // barrier counter
  float* hbuf   = (float*)((char*)d_ws + 256);            // 2 * B*H floats

  const size_t shmem =
      (size_t)(NCOL * WSTRIDE + 2 * BATCH * ASTRIDE + BATCH * GSTRIDE) * sizeof(float);

  // ~208 KB dynamic LDS per WG (ISA allows up to 320 KB per workgroup)
  (void)hipFuncSetAttribute((const void*)lstm_persistent_kernel,
                            hipFuncAttributeMaxDynamicSharedMemorySize, (int)shmem);

  hipLaunchKernelGGL(lstm_init_kernel, dim3(1), dim3(1), 0, stream, cnt);
  hipLaunchKernelGGL(lstm_persistent_kernel, dim3(NWG), dim3(NTHR), shmem, stream,
                     x, h0, c0, Wi, bi, Wh, bh, out, hbuf, cnt);
}